// Encoder_44753559224641
// MI455X (gfx1250) — compile-verified
//
#include <hip/hip_runtime.h>
#include <math.h>

#define Nn 4096
#define Ee 16384
#define FNn 16
#define FEe 8
#define Ll 128
#define Gg 64

typedef __attribute__((ext_vector_type(2))) float v2f;
typedef __attribute__((ext_vector_type(8))) float v8f;

// ---------------------------------------------------------------------------
// degree / norm
// ---------------------------------------------------------------------------
__global__ void zero_kernel(float* __restrict__ p, int n) {
    int i = blockIdx.x * blockDim.x + threadIdx.x;
    if (i < n) p[i] = 0.0f;
}

__global__ void deg_kernel(const int* __restrict__ src, float* __restrict__ deg, int e) {
    int i = blockIdx.x * blockDim.x + threadIdx.x;
    if (i < e) atomicAdd(&deg[src[i]], 1.0f);
}

__global__ void norm_kernel(const float* __restrict__ deg, float* __restrict__ norm, int n) {
    int i = blockIdx.x * blockDim.x + threadIdx.x;
    if (i < n) norm[i] = (deg[i] > 0.0f) ? (1.0f / deg[i]) : 0.0f;
}

// ---------------------------------------------------------------------------
// Repack W2[kk, o] = (kp<8 ? We[kp, o*CIN+ci] : be[o*CIN+ci]),  kk = kp*CIN+ci,
// into [K x COUT] row-major so WMMA B-fragment loads are lane-coalesced.
// ---------------------------------------------------------------------------
template <int CIN, int COUT>
__global__ void w2_pack_kernel(const float* __restrict__ We, const float* __restrict__ be,
                               float* __restrict__ w2) {
    constexpr int K = 9 * CIN;
    int idx = blockIdx.x * blockDim.x + threadIdx.x;
    if (idx >= K * COUT) return;
    int kk = idx / COUT, o = idx % COUT;
    int kp = kk / CIN, ci = kk % CIN;
    w2[idx] = (kp < 8) ? We[kp * (CIN * COUT) + o * CIN + ci] : be[o * CIN + ci];
}

// ---------------------------------------------------------------------------
// WMMA residual GEMM: h_out[v,o] = bias[o] + br[o] + sum_i h_in[v,i]*Wr[i,o]
// One block = 16-node row tile; wave w owns output columns [16w, 16w+16).
// ---------------------------------------------------------------------------
template <int CIN, int COUT>
__global__ __launch_bounds__(32 * (COUT / 16))
void init_out_wmma_kernel(const float* __restrict__ h_in,
                          const float* __restrict__ Wr,
                          const float* __restrict__ br,
                          const float* __restrict__ bias,
                          float* __restrict__ h_out) {
    __shared__ float s_a[16][CIN];
    const int v0 = blockIdx.x * 16;
    const int tid = (int)threadIdx.x;
    const int nthr = (int)blockDim.x;
    for (int i = tid; i < 16 * CIN; i += nthr) {
        int m = i / CIN, c = i % CIN;
        s_a[m][c] = h_in[(v0 + m) * CIN + c];
    }
    __syncthreads();

    const int lane = tid & 31;
    const int n0 = (tid >> 5) * 16;
    const int nn = lane & 15;
    const int khalf = lane >> 4;
    const int o = n0 + nn;

    v8f acc = {0.f, 0.f, 0.f, 0.f, 0.f, 0.f, 0.f, 0.f};
    for (int k = 0; k < CIN; k += 4) {
        const int ka = k + khalf * 2;
        v2f a, b;
        a.x = s_a[nn][ka];
        a.y = s_a[nn][ka + 1];
        b.x = Wr[ka * COUT + o];
        b.y = Wr[(ka + 1) * COUT + o];
        acc = __builtin_amdgcn_wmma_f32_16x16x4_f32(false, a, false, b, (short)0, acc,
                                                    false, false);
    }
    const float cb = bias[o] + br[o];
    const int mbase = khalf * 8;
#pragma unroll
    for (int r = 0; r < 8; ++r) {
        h_out[(v0 + mbase + r) * COUT + o] = acc[r] + cb;
    }
}

// ---------------------------------------------------------------------------
// Fused ECC message GEMM on WMMA (fp32, 16x16x4):
//   msg[e,o] = norm[src_e] * sum_kk Z[e,kk] * W2[kk,o]
//   Z[e, kp*CIN+i] = ea'[e,kp] * x[src_e, i]   (ea' = [edge_attr row, 1.0])
// scatter: atomicAdd into h_out[dst_e, o]
// ---------------------------------------------------------------------------
template <int CIN, int COUT>
__global__ __launch_bounds__(32 * (COUT / 16))
void ecc_gemm_kernel(const float* __restrict__ h_in,
                     const float* __restrict__ ea,
                     const int* __restrict__ src,
                     const int* __restrict__ dst,
                     const float* __restrict__ norm,
                     const float* __restrict__ w2,   // [9*CIN x COUT] packed
                     float* __restrict__ h_out) {
    constexpr int K = 9 * CIN;
    __shared__ float s_ea[16][9];
    __shared__ float s_x[16][CIN];
    __shared__ int s_src[16];
    __shared__ int s_dst[16];

    const int e0 = blockIdx.x * 16;
    const int tid = (int)threadIdx.x;
    const int nthr = (int)blockDim.x;

    for (int i = tid; i < 16; i += nthr) {
        s_src[i] = src[e0 + i];
        s_dst[i] = dst[e0 + i];
    }
    __syncthreads();
    for (int i = tid; i < 16 * 9; i += nthr) {
        int m = i / 9, k = i % 9;
        s_ea[m][k] = (k < 8) ? ea[(e0 + m) * FEe + k] : 1.0f;
    }
    for (int i = tid; i < 16 * CIN; i += nthr) {
        int m = i / CIN, c = i % CIN;
        s_x[m][c] = h_in[s_src[m] * CIN + c];
    }
    __syncthreads();

    const int lane = tid & 31;
    const int n0 = (tid >> 5) * 16;
    const int nn = lane & 15;
    const int khalf = lane >> 4;
    const int o = n0 + nn;

    v8f acc = {0.f, 0.f, 0.f, 0.f, 0.f, 0.f, 0.f, 0.f};
    for (int k = 0; k < K; k += 4) {
        const int ka = k + khalf * 2;
        v2f a, b;
        a.x = s_ea[nn][ka / CIN] * s_x[nn][ka % CIN];
        a.y = s_ea[nn][(ka + 1) / CIN] * s_x[nn][(ka + 1) % CIN];
        b.x = w2[ka * COUT + o];
        b.y = w2[(ka + 1) * COUT + o];
        acc = __builtin_amdgcn_wmma_f32_16x16x4_f32(false, a, false, b, (short)0, acc,
                                                    false, false);
    }

    const int mbase = khalf * 8;
#pragma unroll
    for (int r = 0; r < 8; ++r) {
        const int mm = mbase + r;
        const float val = acc[r] * norm[s_src[mm]];
        atomicAdd(&h_out[s_dst[mm] * COUT + o], val);
    }
}

// ---------------------------------------------------------------------------
// BatchNorm stats (biased variance) : one block per channel
// ---------------------------------------------------------------------------
template <int COUT>
__global__ void bn_stats_kernel(const float* __restrict__ h, float* __restrict__ mv) {
    __shared__ float ss[256];
    __shared__ float sq[256];
    const int o = blockIdx.x;
    float s = 0.f, q = 0.f;
    for (int v = (int)threadIdx.x; v < Nn; v += 256) {
        float x = h[v * COUT + o];
        s += x;
        q += x * x;
    }
    ss[threadIdx.x] = s;
    sq[threadIdx.x] = q;
    __syncthreads();
    for (int w = 128; w > 0; w >>= 1) {
        if ((int)threadIdx.x < w) {
            ss[threadIdx.x] += ss[threadIdx.x + w];
            sq[threadIdx.x] += sq[threadIdx.x + w];
        }
        __syncthreads();
    }
    if (threadIdx.x == 0) {
        float m = ss[0] / (float)Nn;
        mv[o] = m;
        mv[COUT + o] = sq[0] / (float)Nn - m * m;
    }
}

template <int COUT>
__global__ void bn_prelu_kernel(float* __restrict__ h, const float* __restrict__ mv,
                                const float* __restrict__ g, const float* __restrict__ b,
                                const float* __restrict__ a) {
    int idx = blockIdx.x * blockDim.x + threadIdx.x;
    if (idx >= Nn * COUT) return;
    int o = idx % COUT;
    float m = mv[o], var = mv[COUT + o];
    float v = (h[idx] - m) * rsqrtf(var + 1e-5f) * g[o] + b[o];
    h[idx] = (v >= 0.f) ? v : a[0] * v;
}

// ---------------------------------------------------------------------------
// Fused gated-pooling GEMMs on WMMA: for a 16-node row tile, compute
//   Gi = h@Wi (accG) and Fj = h@Wj (accF) sharing one staged A tile,
// then pooled[batch[v], o] += sigmoid(Gi+bi)*tanh(Fj+bj) straight from C regs.
// 8 waves per block cover the 128 output columns.
// ---------------------------------------------------------------------------
__global__ __launch_bounds__(256)
void pool_wmma_kernel(const float* __restrict__ h,
                      const float* __restrict__ Wi, const float* __restrict__ bi,
                      const float* __restrict__ Wj, const float* __restrict__ bj,
                      const int* __restrict__ batch, float* __restrict__ pooled) {
    __shared__ float s_h[16][Ll];
    __shared__ int s_b[16];
    const int v0 = blockIdx.x * 16;
    const int tid = (int)threadIdx.x;

    for (int i = tid; i < 16; i += 256) s_b[i] = batch[v0 + i];
    for (int i = tid; i < 16 * Ll; i += 256) {
        int m = i / Ll, c = i % Ll;
        s_h[m][c] = h[(v0 + m) * Ll + c];
    }
    __syncthreads();

    const int lane = tid & 31;
    const int n0 = (tid >> 5) * 16;
    const int nn = lane & 15;
    const int khalf = lane >> 4;
    const int o = n0 + nn;

    v8f accG = {0.f, 0.f, 0.f, 0.f, 0.f, 0.f, 0.f, 0.f};
    v8f accF = {0.f, 0.f, 0.f, 0.f, 0.f, 0.f, 0.f, 0.f};
    for (int k = 0; k < Ll; k += 4) {
        const int ka = k + khalf * 2;
        v2f a, bg, bf;
        a.x = s_h[nn][ka];
        a.y = s_h[nn][ka + 1];
        bg.x = Wi[ka * Ll + o];
        bg.y = Wi[(ka + 1) * Ll + o];
        bf.x = Wj[ka * Ll + o];
        bf.y = Wj[(ka + 1) * Ll + o];
        accG = __builtin_amdgcn_wmma_f32_16x16x4_f32(false, a, false, bg, (short)0, accG,
                                                     false, false);
        accF = __builtin_amdgcn_wmma_f32_16x16x4_f32(false, a, false, bf, (short)0, accF,
                                                     false, false);
    }

    const float cbi = bi[o], cbj = bj[o];
    const int mbase = khalf * 8;
#pragma unroll
    for (int r = 0; r < 8; ++r) {
        const int mm = mbase + r;
        float gi = accG[r] + cbi;
        float fj = accF[r] + cbj;
        float val = (1.0f / (1.0f + expf(-gi))) * tanhf(fj);
        atomicAdd(&pooled[s_b[mm] * Ll + o], val);
    }
}

// z[g,c] = bfc[c] + sum_k tanh(pooled[g,k]) * Wfc[k,c]; split into (mu, logvar)
__global__ void final_kernel(const float* __restrict__ pooled,
                             const float* __restrict__ Wfc, const float* __restrict__ bfc,
                             float* __restrict__ out) {
    int idx = blockIdx.x * blockDim.x + threadIdx.x;
    if (idx >= Gg * 2 * Ll) return;
    int g = idx / (2 * Ll), c = idx % (2 * Ll);
    float acc = bfc[c];
#pragma unroll 4
    for (int k = 0; k < Ll; ++k) acc += tanhf(pooled[g * Ll + k]) * Wfc[k * (2 * Ll) + c];
    if (c < Ll)
        out[g * Ll + c] = acc;                  // first output  z[:, :L]
    else
        out[Gg * Ll + g * Ll + (c - Ll)] = acc; // second output z[:, L:]
}

// ---------------------------------------------------------------------------
extern "C" void kernel_launch(void* const* d_in, const int* in_sizes, int n_in,
                              void* d_out, int out_size, void* d_ws, size_t ws_size,
                              hipStream_t stream) {
    const float* x  = (const float*)d_in[0];
    const float* ea = (const float*)d_in[1];
    const int* eidx = (const int*)d_in[2];
    const int* batch = (const int*)d_in[3];
    const int* src = eidx;
    const int* dst = eidx + Ee;

    const float* We[3];  const float* be[3];  const float* bias[3];
    const float* Wr[3];  const float* br[3];  const float* bg[3];
    const float* bb[3];  const float* al[3];
    for (int l = 0; l < 3; ++l) {
        int base = 4 + 8 * l;
        We[l]   = (const float*)d_in[base + 0];
        be[l]   = (const float*)d_in[base + 1];
        bias[l] = (const float*)d_in[base + 2];
        Wr[l]   = (const float*)d_in[base + 3];
        br[l]   = (const float*)d_in[base + 4];
        bg[l]   = (const float*)d_in[base + 5];
        bb[l]   = (const float*)d_in[base + 6];
        al[l]   = (const float*)d_in[base + 7];
    }
    const float* Wi  = (const float*)d_in[28];
    const float* bi  = (const float*)d_in[29];
    const float* Wj  = (const float*)d_in[30];
    const float* bj  = (const float*)d_in[31];
    const float* Wfc = (const float*)d_in[32];
    const float* bfc = (const float*)d_in[33];
    float* out = (float*)d_out;

    // workspace layout (floats)
    float* deg    = (float*)d_ws;
    float* norm   = deg + Nn;
    float* h1     = norm + Nn;            // N x 32
    float* h2     = h1 + Nn * 32;         // N x 64
    float* h3     = h2 + Nn * 64;         // N x 128
    float* mv     = h3 + Nn * 128;        // 2 x 128 (reused per layer)
    float* pooled = mv + 256;             // G x 128
    float* w2a    = pooled + Gg * Ll;     // 144 x 32
    float* w2b    = w2a + 144 * 32;       // 288 x 64
    float* w2c    = w2b + 288 * 64;       // 576 x 128

    // degree / norm / packed weights
    zero_kernel<<<(Nn + 255) / 256, 256, 0, stream>>>(deg, Nn);
    zero_kernel<<<(Gg * Ll + 255) / 256, 256, 0, stream>>>(pooled, Gg * Ll);
    deg_kernel<<<Ee / 256, 256, 0, stream>>>(src, deg, Ee);
    norm_kernel<<<Nn / 256, 256, 0, stream>>>(deg, norm, Nn);
    w2_pack_kernel<16, 32><<<(144 * 32 + 255) / 256, 256, 0, stream>>>(We[0], be[0], w2a);
    w2_pack_kernel<32, 64><<<(288 * 64 + 255) / 256, 256, 0, stream>>>(We[1], be[1], w2b);
    w2_pack_kernel<64, 128><<<(576 * 128 + 255) / 256, 256, 0, stream>>>(We[2], be[2], w2c);

    // ---- layer 1: 16 -> 32 ----
    init_out_wmma_kernel<16, 32><<<Nn / 16, 32 * (32 / 16), 0, stream>>>(x, Wr[0], br[0],
                                                                         bias[0], h1);
    ecc_gemm_kernel<16, 32><<<Ee / 16, 32 * (32 / 16), 0, stream>>>(x, ea, src, dst, norm,
                                                                    w2a, h1);
    bn_stats_kernel<32><<<32, 256, 0, stream>>>(h1, mv);
    bn_prelu_kernel<32><<<(Nn * 32) / 256, 256, 0, stream>>>(h1, mv, bg[0], bb[0], al[0]);

    // ---- layer 2: 32 -> 64 ----
    init_out_wmma_kernel<32, 64><<<Nn / 16, 32 * (64 / 16), 0, stream>>>(h1, Wr[1], br[1],
                                                                         bias[1], h2);
    ecc_gemm_kernel<32, 64><<<Ee / 16, 32 * (64 / 16), 0, stream>>>(h1, ea, src, dst, norm,
                                                                    w2b, h2);
    bn_stats_kernel<64><<<64, 256, 0, stream>>>(h2, mv);
    bn_prelu_kernel<64><<<(Nn * 64) / 256, 256, 0, stream>>>(h2, mv, bg[1], bb[1], al[1]);

    // ---- layer 3: 64 -> 128 ----
    init_out_wmma_kernel<64, 128><<<Nn / 16, 32 * (128 / 16), 0, stream>>>(h2, Wr[2], br[2],
                                                                           bias[2], h3);
    ecc_gemm_kernel<64, 128><<<Ee / 16, 32 * (128 / 16), 0, stream>>>(h2, ea, src, dst, norm,
                                                                      w2c, h3);
    bn_stats_kernel<128><<<128, 256, 0, stream>>>(h3, mv);
    bn_prelu_kernel<128><<<(Nn * 128) / 256, 256, 0, stream>>>(h3, mv, bg[2], bb[2], al[2]);

    // ---- pooling + head ----
    pool_wmma_kernel<<<Nn / 16, 256, 0, stream>>>(h3, Wi, bi, Wj, bj, batch, pooled);
    final_kernel<<<(Gg * 2 * Ll) / 256, 256, 0, stream>>>(pooled, Wfc, bfc, out);
}